// VesselGNN_2D_36077725286937
// MI455X (gfx1250) — compile-verified
//
#include <hip/hip_runtime.h>
#include <hip/hip_bf16.h>
#include <math.h>

#define N_NODES 100000
#define N_EDGES 3200000
#define N_GRAPHS 64
#define IN_DIM 4
#define HID_DIM 64
#define EMB_DIM 32

typedef __attribute__((ext_vector_type(2))) float v2f;
typedef __attribute__((ext_vector_type(8))) float v8f;

// ---------------------------------------------------------------------------
// Degree / normalization
// ---------------------------------------------------------------------------
__global__ void deg_kernel(const int* __restrict__ dst, float* __restrict__ deg, int E) {
    int t = blockIdx.x * blockDim.x + threadIdx.x;
    if (t < E) atomicAdd(&deg[dst[t]], 1.0f);
}

__global__ void dis_kernel(const float* __restrict__ deg, float* __restrict__ dis, int N) {
    int t = blockIdx.x * blockDim.x + threadIdx.x;
    if (t < N) dis[t] = rsqrtf(deg[t] + 1.0f);   // self-loop included, deg+1 >= 1
}

// ---------------------------------------------------------------------------
// Dense GEMM  H[N,F] = X[N,K] * W[K,F]  via V_WMMA_F32_16X16X4_F32 (wave32).
// K, F compile-time: fully unrolled K-chain of WMMAs, constant address math.
// One wave computes a 16x16 tile; wave w of the block owns feature tile w.
// A layout: lane (k>=2)*16+m holds x[row0+m][kk + 2*(lane/16) + v] in vgpr v.
// B layout: lane (k>=2)*16+n holds W[kk + 2*(lane/16) + v][n0+n] in vgpr v.
// D layout: vgpr j -> row = row0 + j + 8*(lane/16), col = n0 + lane%16.
// K multiple of 4, N multiple of 16, EXEC all ones (no divergence).
// ---------------------------------------------------------------------------
template <int K, int F>
__global__ __launch_bounds__(32 * (F / 16)) void gemm_wmma_f32(
        const float* __restrict__ X, const float* __restrict__ W,
        float* __restrict__ H) {
    const int lane = threadIdx.x & 31;
    const int n0   = (threadIdx.x >> 5) * 16;     // feature tile of this wave
    const int row0 = blockIdx.x * 16;
    const int g = lane >> 4;                      // lane half: 0 or 1
    const int m = lane & 15;                      // row within A tile / col within B,D

    v8f acc = {};                                 // C = 0

    const float* ap = X + (size_t)(row0 + m) * K + 2 * g;
    const float* bp = W + (size_t)(2 * g) * F + n0 + m;

#pragma unroll
    for (int kk = 0; kk < K; kk += 4) {
        v2f a = *(const v2f*)(ap + kk);           // k = kk+2g, kk+2g+1
        v2f b;
        b.x = bp[kk * F];                         // k = kk+2g,   col n0+m
        b.y = bp[(kk + 1) * F];                   // k = kk+2g+1, col n0+m
        acc = __builtin_amdgcn_wmma_f32_16x16x4_f32(
            /*neg_a=*/false, a, /*neg_b=*/false, b,
            /*c_mod=*/(short)0, acc, /*reuse_a=*/false, /*reuse_b=*/false);
    }

    float* hp = H + (size_t)row0 * F + n0 + m;
#pragma unroll
    for (int j = 0; j < 8; ++j)
        hp[(j + 8 * g) * F] = acc[j];
}

// ---------------------------------------------------------------------------
// Edge scatter: AGG[dst] += dis[src]*dis[dst] * H[src].
// F/4 lanes per edge, float4 per lane -> 256B/128B contiguous gather per
// edge; atomics resolve in the 192MB L2 (the whole H array fits: 25.6MB).
// F compile-time so tid / (F/4) and tid % (F/4) are shift/mask.
// ---------------------------------------------------------------------------
template <int F>
__global__ void aggregate_edges(const float* __restrict__ H, const int* __restrict__ src,
                                const int* __restrict__ dst, const float* __restrict__ dis,
                                float* __restrict__ AGG, long long total) {
    constexpr int F4 = F / 4;
    long long tid = (long long)blockIdx.x * blockDim.x + threadIdx.x;
    if (tid >= total) return;
    int e = (int)(tid / F4);
    int c = (int)(tid % F4);
    int s = src[e], d = dst[e];
    float coef = dis[s] * dis[d];
    float4 v = *((const float4*)(H + (size_t)s * F) + c);
    float* ap = AGG + (size_t)d * F + 4 * c;
    atomicAdd(ap + 0, coef * v.x);
    atomicAdd(ap + 1, coef * v.y);
    atomicAdd(ap + 2, coef * v.z);
    atomicAdd(ap + 3, coef * v.w);
}

// ---------------------------------------------------------------------------
// Finalize: AGG += dis^2 * H + bias, optional ReLU (in place).
// ---------------------------------------------------------------------------
template <int F, bool RELU>
__global__ void finalize_nodes(const float* __restrict__ H, float* __restrict__ AGG,
                               const float* __restrict__ dis, const float* __restrict__ bias,
                               long long total) {
    long long tid = (long long)blockIdx.x * blockDim.x + threadIdx.x;
    if (tid >= total) return;
    int i = (int)(tid / F);
    int f = (int)(tid % F);
    float di = dis[i];
    float v = AGG[tid] + di * di * H[tid] + bias[f];
    if (RELU) v = fmaxf(v, 0.0f);
    AGG[tid] = v;
}

// ---------------------------------------------------------------------------
// Graph pooling (mean + max).
// ---------------------------------------------------------------------------
__device__ __forceinline__ void atomicMaxF(float* addr, float v) {
    // Monotonic int trick: valid with -inf init, single hardware atomic.
    if (v >= 0.0f) atomicMax((int*)addr, __float_as_int(v));
    else           atomicMin((unsigned int*)addr, (unsigned int)__float_as_int(v));
}

__global__ void pool_init(float* __restrict__ psum, float* __restrict__ pmax,
                          float* __restrict__ pcnt) {
    int t = blockIdx.x * blockDim.x + threadIdx.x;
    if (t < N_GRAPHS * EMB_DIM) { psum[t] = 0.0f; pmax[t] = -INFINITY; }
    if (t < N_GRAPHS) pcnt[t] = 0.0f;
}

__global__ void pool_kernel(const float* __restrict__ emb, const int* __restrict__ batch,
                            float* __restrict__ psum, float* __restrict__ pmax,
                            float* __restrict__ pcnt, long long total) {
    long long tid = (long long)blockIdx.x * blockDim.x + threadIdx.x;
    if (tid >= total) return;
    int i = (int)(tid / EMB_DIM);
    int f = (int)(tid % EMB_DIM);
    int gph = batch[i];
    float v = emb[tid];
    atomicAdd(&psum[gph * EMB_DIM + f], v);
    atomicMaxF(&pmax[gph * EMB_DIM + f], v);
    if (f == 0) atomicAdd(&pcnt[gph], 1.0f);
}

__global__ void pool_out(const float* __restrict__ psum, const float* __restrict__ pmax,
                         const float* __restrict__ pcnt, float* __restrict__ out) {
    int t = blockIdx.x * blockDim.x + threadIdx.x;
    if (t >= N_GRAPHS * EMB_DIM) return;
    int gph = t / EMB_DIM;
    int f   = t % EMB_DIM;
    float c = fmaxf(pcnt[gph], 1.0f);
    out[gph * (2 * EMB_DIM) + f]           = psum[t] / c;
    out[gph * (2 * EMB_DIM) + EMB_DIM + f] = pmax[t];
}

// ---------------------------------------------------------------------------
// Host-side orchestration
// ---------------------------------------------------------------------------
static inline int cdiv_ll(long long a, int b) { return (int)((a + b - 1) / b); }

extern "C" void kernel_launch(void* const* d_in, const int* in_sizes, int n_in,
                              void* d_out, int out_size, void* d_ws, size_t ws_size,
                              hipStream_t stream) {
    const float* x   = (const float*)d_in[0];
    const float* W1  = (const float*)d_in[1];
    const float* b1  = (const float*)d_in[2];
    const float* W2  = (const float*)d_in[3];
    const float* b2  = (const float*)d_in[4];
    const float* W3  = (const float*)d_in[5];
    const float* b3  = (const float*)d_in[6];
    const int*   ei  = (const int*)d_in[7];
    const int*   bat = (const int*)d_in[8];
    float* out = (float*)d_out;

    const int N = N_NODES, E = N_EDGES;
    const int* src = ei;
    const int* dst = ei + E;

    // Workspace layout (floats)
    float* deg  = (float*)d_ws;                          // N
    float* dis  = deg  + N;                              // N
    float* hbuf = dis  + N;                              // N * 64
    float* abuf = hbuf + (size_t)N * HID_DIM;            // N * 64
    float* psum = abuf + (size_t)N * HID_DIM;            // 64*32
    float* pmax = psum + N_GRAPHS * EMB_DIM;             // 64*32
    float* pcnt = pmax + N_GRAPHS * EMB_DIM;             // 64

    // ---- degrees & normalization --------------------------------------
    hipMemsetAsync(deg, 0, (size_t)N * sizeof(float), stream);
    deg_kernel<<<cdiv_ll(E, 256), 256, 0, stream>>>(dst, deg, E);
    dis_kernel<<<cdiv_ll(N, 256), 256, 0, stream>>>(deg, dis, N);

    const int row_tiles = N / 16;   // 6250, exact

    // ---- layer 1: x[N,4] -> abuf[N,64], ReLU --------------------------
    gemm_wmma_f32<IN_DIM, HID_DIM>
        <<<row_tiles, 32 * (HID_DIM / 16), 0, stream>>>(x, W1, hbuf);
    hipMemsetAsync(abuf, 0, (size_t)N * HID_DIM * sizeof(float), stream);
    {
        long long tot = (long long)E * (HID_DIM / 4);
        aggregate_edges<HID_DIM>
            <<<cdiv_ll(tot, 256), 256, 0, stream>>>(hbuf, src, dst, dis, abuf, tot);
        long long ntot = (long long)N * HID_DIM;
        finalize_nodes<HID_DIM, true>
            <<<cdiv_ll(ntot, 256), 256, 0, stream>>>(hbuf, abuf, dis, b1, ntot);
    }

    // ---- layer 2: abuf[N,64] -> abuf[N,64], ReLU ----------------------
    gemm_wmma_f32<HID_DIM, HID_DIM>
        <<<row_tiles, 32 * (HID_DIM / 16), 0, stream>>>(abuf, W2, hbuf);
    hipMemsetAsync(abuf, 0, (size_t)N * HID_DIM * sizeof(float), stream);
    {
        long long tot = (long long)E * (HID_DIM / 4);
        aggregate_edges<HID_DIM>
            <<<cdiv_ll(tot, 256), 256, 0, stream>>>(hbuf, src, dst, dis, abuf, tot);
        long long ntot = (long long)N * HID_DIM;
        finalize_nodes<HID_DIM, true>
            <<<cdiv_ll(ntot, 256), 256, 0, stream>>>(hbuf, abuf, dis, b2, ntot);
    }

    // ---- layer 3: abuf[N,64] -> abuf[N,32], no ReLU -------------------
    gemm_wmma_f32<HID_DIM, EMB_DIM>
        <<<row_tiles, 32 * (EMB_DIM / 16), 0, stream>>>(abuf, W3, hbuf);
    hipMemsetAsync(abuf, 0, (size_t)N * EMB_DIM * sizeof(float), stream);
    {
        long long tot = (long long)E * (EMB_DIM / 4);
        aggregate_edges<EMB_DIM>
            <<<cdiv_ll(tot, 256), 256, 0, stream>>>(hbuf, src, dst, dis, abuf, tot);
        long long ntot = (long long)N * EMB_DIM;
        finalize_nodes<EMB_DIM, false>
            <<<cdiv_ll(ntot, 256), 256, 0, stream>>>(hbuf, abuf, dis, b3, ntot);
    }

    // ---- pooling ------------------------------------------------------
    pool_init<<<cdiv_ll(N_GRAPHS * EMB_DIM, 256), 256, 0, stream>>>(psum, pmax, pcnt);
    {
        long long tot = (long long)N * EMB_DIM;
        pool_kernel<<<cdiv_ll(tot, 256), 256, 0, stream>>>(abuf, bat, psum, pmax, pcnt, tot);
    }
    pool_out<<<cdiv_ll(N_GRAPHS * EMB_DIM, 256), 256, 0, stream>>>(psum, pmax, pcnt, out);
}